// DownsamplingBlock_68384469287499
// MI455X (gfx1250) — compile-verified
//
#include <hip/hip_runtime.h>

typedef float v2f __attribute__((ext_vector_type(2)));
typedef float v8f __attribute__((ext_vector_type(8)));

#define C_IN      96
#define C_OUT     192
#define NBATCH    8
#define BLK_M     64     // pooled rows (windows) per block
#define AS_STRIDE 98     // LDS row stride (words) for A tile, bank-friendly, 8B aligned
#define NTILE     12     // 12*16 = 192: full output width per block (feat read ONCE)
#define NKG       (2 * (C_IN / 4))   // 48 K-pair groups in packed W

// Map window id m -> (start row in feat, count 1..4, batch index) from the 8
// ragged batch lengths. 8-iteration prefix scan; lengths stays hot in cache.
__device__ __forceinline__ void window_map(const int* __restrict__ lengths, int m,
                                           long* start, int* cnt, int* bsel) {
  long off = 0, woff = 0;
  int  b = -1;  long st = 0;  int c = 1;
  #pragma unroll
  for (int bi = 0; bi < NBATCH; ++bi) {
    long L  = (long)lengths[bi];
    long nw = (L + 3) >> 2;
    if (b < 0 && (long)m < woff + nw) {
      b = bi;
      long p   = ((long)m - woff) * 4;
      st       = off + p;
      long rem = L - p;
      c = rem < 4 ? (int)rem : 4;
    }
    off += L;  woff += nw;
  }
  *start = st;  *cnt = c;  *bsel = b;
}

// Fused: windowed mean-pool of feat (into LDS) + fp32 WMMA GEMM (96 -> 192) + bias + ReLU.
// Grid: ceil(M/64). Block: 128 threads = 4 waves; wave w owns M-tile rows [w*16, w*16+16),
// each wave computes the full 192 output columns (12 WMMA N-tiles), so feat is read once.
__global__ __launch_bounds__(128) void pool_gemm_kernel(
    const float* __restrict__ feat, const float* __restrict__ W,
    const float* __restrict__ bias, const int* __restrict__ lengths,
    float* __restrict__ out, int M) {
  __shared__ float As[BLK_M * AS_STRIDE];   // pooled 64 x 96 tile        (25088 B)
  __shared__ v2f   Wp[NKG * C_OUT];         // K-pair-packed W fragments  (73728 B)

  const int tid       = threadIdx.x;
  const int mbase_blk = blockIdx.x * BLK_M;

  // ---- stage W into LDS, packed as B fragments: Wp[kk*2+g][n] = (W[4kk+2g][n], W[4kk+2g+1][n])
  // so each inner-loop B fragment is a single aligned ds_load_b64 (no repacking movs).
  for (int i = tid; i < NKG * (C_OUT / 2); i += 128) {
    const int kkg = i / (C_OUT / 2);
    const int n2  = (i % (C_OUT / 2)) * 2;
    const int k0  = (kkg >> 1) * 4 + (kkg & 1) * 2;
    const v2f r0 = *(const v2f*)(W + (size_t)(k0    ) * C_OUT + n2);  // coalesced b64
    const v2f r1 = *(const v2f*)(W + (size_t)(k0 + 1) * C_OUT + n2);  // coalesced b64
    Wp[kkg * C_OUT + n2    ] = (v2f){r0.x, r1.x};
    Wp[kkg * C_OUT + n2 + 1] = (v2f){r0.y, r1.y};
  }

  // ---- windowed mean-pool: 64 windows x 96 channels into LDS ----
  const int lane = tid & 31;
  const int wgrp = tid >> 5;                // 4 window-slots in flight
  for (int wl = wgrp; wl < BLK_M; wl += 4) {
    const int m = mbase_blk + wl;           // uniform per wave -> no lane divergence
    if (m < M) {
      long start; int cnt, bsel;
      window_map(lengths, m, &start, &cnt, &bsel);
      const float inv = 1.0f / (float)cnt;
      for (int c = lane; c < C_IN; c += 32) {
        float s = 0.0f;
        for (int r = 0; r < cnt; ++r)
          s += feat[(size_t)(start + r) * C_IN + c];
        As[wl * AS_STRIDE + c] = s * inv;
      }
      // prefetch next block's feat rows (gfx1250 global_prefetch path)
      if (m + BLK_M < M) {
        const char* p = (const char*)(feat + (size_t)(start + 4 * BLK_M) * C_IN);
        __builtin_prefetch(p + lane * 128, 0, 1);
      }
    } else {
      for (int c = lane; c < C_IN; c += 32)
        As[wl * AS_STRIDE + c] = 0.0f;
    }
  }
  __syncthreads();

  // ---- WMMA: each wave computes 16(M) x 192(N), K = 96 in steps of 4 ----
  const int wave  = tid >> 5;
  const int hlane = lane & 15;
  const int kadd  = (lane < 16) ? 0 : 2;    // A/B fragment K-offset per ISA layout
  const int arow  = wave * 16 + hlane;

  v8f acc[NTILE] = {};
  for (int kk = 0; kk < C_IN / 4; ++kk) {
    const v2f a = *(const v2f*)(As + arow * AS_STRIDE + 4 * kk + kadd);  // b64 ds load
    const v2f* brow = Wp + (kk * 2 + (kadd >> 1)) * C_OUT;
    #pragma unroll
    for (int t = 0; t < NTILE; ++t) {
      const v2f bf = brow[t * 16 + hlane];                               // b64 ds load
      acc[t] = __builtin_amdgcn_wmma_f32_16x16x4_f32(
          false, a, false, bf, (short)0, acc[t], false, false);
    }
  }

  // ---- epilogue: bias + ReLU. C layout: VGPR r -> M=r (lanes 0-15) / r+8 (16-31) ----
  const int mrow0 = mbase_blk + wave * 16 + ((lane < 16) ? 0 : 8);
  if (mbase_blk + BLK_M <= M) {             // fast path: no per-element guards
    #pragma unroll
    for (int t = 0; t < NTILE; ++t) {
      const int   col = t * 16 + hlane;
      const float bv  = bias[col];
      #pragma unroll
      for (int r = 0; r < 8; ++r) {
        const float v = acc[t][r] + bv;
        out[(size_t)(mrow0 + r) * C_OUT + col] = v > 0.0f ? v : 0.0f;
      }
    }
  } else {
    #pragma unroll
    for (int t = 0; t < NTILE; ++t) {
      const int   col = t * 16 + hlane;
      const float bv  = bias[col];
      #pragma unroll
      for (int r = 0; r < 8; ++r) {
        const int m = mrow0 + r;
        if (m < M) {
          const float v = acc[t][r] + bv;
          out[(size_t)m * C_OUT + col] = v > 0.0f ? v : 0.0f;
        }
      }
    }
  }
}

// Small outputs: coord/scores means, batch ids, lengths_down, offset_down.
__global__ __launch_bounds__(256) void pool_aux_kernel(
    const float* __restrict__ coord, const float* __restrict__ scores,
    const int* __restrict__ lengths,
    float* __restrict__ coord_down, float* __restrict__ batch_down,
    float* __restrict__ scores_down, float* __restrict__ lengths_down,
    float* __restrict__ offset_down, int M) {
  const int m = blockIdx.x * blockDim.x + threadIdx.x;
  if (m < M) {
    long start; int cnt, bsel;
    window_map(lengths, m, &start, &cnt, &bsel);
    const float inv = 1.0f / (float)cnt;
    float cx = 0.f, cy = 0.f, cz = 0.f, sc = 0.f;
    for (int r = 0; r < cnt; ++r) {
      const float* cp = coord + (size_t)(start + r) * 3;
      cx += cp[0];  cy += cp[1];  cz += cp[2];
      sc += scores[start + r];
    }
    coord_down[(size_t)m * 3 + 0] = cx * inv;
    coord_down[(size_t)m * 3 + 1] = cy * inv;
    coord_down[(size_t)m * 3 + 2] = cz * inv;
    scores_down[m] = sc * inv;
    batch_down[m]  = (float)bsel;
  }
  if (m < NBATCH) {
    long acc = 0;
    for (int bi = 0; bi <= m; ++bi) acc += ((long)lengths[bi] + 3) >> 2;
    lengths_down[m] = (float)(((long)lengths[m] + 3) >> 2);
    offset_down[m]  = (float)acc;
  }
}

extern "C" void kernel_launch(void* const* d_in, const int* in_sizes, int n_in,
                              void* d_out, int out_size, void* d_ws, size_t ws_size,
                              hipStream_t stream) {
  (void)in_sizes; (void)n_in; (void)d_ws; (void)ws_size;
  const float* feat    = (const float*)d_in[0];
  const float* coord   = (const float*)d_in[1];
  const float* scores  = (const float*)d_in[2];
  const float* W       = (const float*)d_in[3];
  const float* bias    = (const float*)d_in[4];
  const int*   lengths = (const int*)d_in[5];
  // d_in[6] = batch ids: unused, derived from lengths on device.

  // out_size = M*(192 + 3 + 1 + 1) + 16  =>  M recoverable on host.
  const int M = (out_size - 2 * NBATCH) / (C_OUT + 3 + 1 + 1);

  float* out_feat   = (float*)d_out;
  float* out_coord  = out_feat   + (size_t)M * C_OUT;
  float* out_batch  = out_coord  + (size_t)M * 3;
  float* out_scores = out_batch  + M;
  float* out_len    = out_scores + M;
  float* out_off    = out_len    + NBATCH;

  pool_gemm_kernel<<<(M + BLK_M - 1) / BLK_M, 128, 0, stream>>>(
      feat, W, bias, lengths, out_feat, M);

  pool_aux_kernel<<<(M + 255) / 256, 256, 0, stream>>>(
      coord, scores, lengths, out_coord, out_batch, out_scores, out_len, out_off, M);
}